// IcoPool_8641474199778
// MI455X (gfx1250) — compile-verified
//
#include <hip/hip_runtime.h>
#include <hip/hip_bf16.h>
#include <stdint.h>

// Problem constants (from the reference)
#define B_DIM   8
#define F_DIM   64
#define V_IN    163842
#define V_OUT   40962
#define K_NB    7

#define TV      256   // vertices per block (= blockDim.x), 8 waves on CDNA5 wave32
#define NBF     8     // (b,f) rows handled per block; index regs reused NBF times

__global__ __launch_bounds__(TV) void
IcoPool_gather_mean_kernel(const float* __restrict__ x,
                           const long long* __restrict__ nidx,
                           float* __restrict__ out)
{
    // LDS staging area for this block's index tile: 256 verts * 7 idx * 8B = 14336 B
    __shared__ long long sIdx[TV * K_NB];

    const int tid  = threadIdx.x;
    const int v    = blockIdx.x * TV + tid;       // output vertex this lane owns
    const int row0 = blockIdx.y * NBF;            // first (b*F+f) row for this block
    const bool valid = (v < V_OUT);

    // ---- CDNA5 async global->LDS staging of this lane's 7 indices ------------
    // Each lane copies its own 56B of neigh_indices into its private LDS slot.
    // global_load_async_to_lds_b64 vdst(lds-addr), vaddr(64b global), off
    // Tracked by ASYNCcnt; per-lane EXEC masking handles the tail block.
    if (valid) {
        const long long* g = nidx + (size_t)v * K_NB;
        // Low 32 bits of a flat shared pointer are the LDS byte offset
        // (shared aperture lives entirely in the upper 32 address bits).
        unsigned lds_base = (unsigned)(uintptr_t)(&sIdx[tid * K_NB]);
        #pragma unroll
        for (int k = 0; k < K_NB; ++k) {
            asm volatile("global_load_async_to_lds_b64 %0, %1, off"
                         :
                         : "v"(lds_base + 8u * (unsigned)k),
                           "v"((const void*)(g + k))
                         : "memory");
        }
    }
#if __has_builtin(__builtin_amdgcn_s_wait_asynccnt)
    __builtin_amdgcn_s_wait_asynccnt(0);
#else
    asm volatile("s_wait_asynccnt 0" ::: "memory");
#endif
    __syncthreads();

    // ---- Pull indices from LDS into registers (reused across NBF rows) -------
    unsigned idx[K_NB];
    if (valid) {
        #pragma unroll
        for (int k = 0; k < K_NB; ++k)
            idx[k] = (unsigned)sIdx[tid * K_NB + k];
    } else {
        #pragma unroll
        for (int k = 0; k < K_NB; ++k) idx[k] = 0u;
    }

    // ---- Gather + mean over NBF rows; NT store (output never re-read) --------
    const float inv7 = 1.0f / 7.0f;
    #pragma unroll
    for (int r = 0; r < NBF; ++r) {
        const int row = row0 + r;                              // row = b*F + f
        const float* __restrict__ xr = x + (size_t)row * V_IN; // uniform saddr base
        if (valid) {
            float s = 0.0f;
            #pragma unroll
            for (int k = 0; k < K_NB; ++k)
                s += xr[idx[k]];
            __builtin_nontemporal_store(s * inv7,
                                        &out[(size_t)row * V_OUT + v]);
        }
    }
}

extern "C" void kernel_launch(void* const* d_in, const int* in_sizes, int n_in,
                              void* d_out, int out_size, void* d_ws, size_t ws_size,
                              hipStream_t stream) {
    (void)in_sizes; (void)n_in; (void)out_size; (void)d_ws; (void)ws_size;
    const float*     x    = (const float*)d_in[0];       // (B, F, V_IN) f32
    const long long* nidx = (const long long*)d_in[1];   // (V_OUT, K)  i64
    float*           out  = (float*)d_out;               // (B, F, V_OUT) f32

    dim3 grid((V_OUT + TV - 1) / TV, (B_DIM * F_DIM) / NBF); // (161, 64)
    dim3 block(TV);
    IcoPool_gather_mean_kernel<<<grid, block, 0, stream>>>(x, nidx, out);
}